// GCN_9345848836220
// MI455X (gfx1250) — compile-verified
//
#include <hip/hip_runtime.h>
#include <hip/hip_bf16.h>

typedef __attribute__((ext_vector_type(2))) float v2f;
typedef __attribute__((ext_vector_type(8))) float v8f;

#define F_IN  128
#define F_HID 128
#define F_OUT 16
#define AS_STRIDE 132   // 128 + 4 pad: rows land on different LDS banks

static __device__ __forceinline__ v8f wmma_f32_k4(v2f a, v2f b, v8f c) {
    // D = A(16x4,f32) * B(4x16,f32) + C(16x16,f32)
    return __builtin_amdgcn_wmma_f32_16x16x4_f32(
        /*neg_a=*/false, a, /*neg_b=*/false, b,
        /*c_mod=*/(short)0, c, /*reuse_a=*/false, /*reuse_b=*/false);
}

// Async cache->LDS DMA (CDNA5 path, tracked on ASYNCcnt). LDS operand is the
// 32-bit LDS offset: per the flat-aperture mapping, low 32 bits of the generic
// address of a __shared__ object are the LDS byte address.
static __device__ __forceinline__ void async_copy_b128_to_lds(void* lds_ptr,
                                                              const void* gptr) {
    unsigned lds_off = (unsigned)(unsigned long long)lds_ptr;
    asm volatile("global_load_async_to_lds_b128 %0, %1, off"
                 :: "v"(lds_off), "v"(gptr)
                 : "memory");
}
static __device__ __forceinline__ void wait_async_all() {
    asm volatile("s_wait_asynccnt 0" ::: "memory");
}

// ---------------------------------------------------------------------------
// Zero an int buffer (degree histogram).
// ---------------------------------------------------------------------------
__global__ __launch_bounds__(256) void gcn_zero_i32(int* p, int n) {
    int i = blockIdx.x * blockDim.x + threadIdx.x;
    int stride = gridDim.x * blockDim.x;
    for (; i < n; i += stride) p[i] = 0;
}

// ---------------------------------------------------------------------------
// CSR build step 1: degree histogram (integer atomics, ~E total).
// ---------------------------------------------------------------------------
__global__ __launch_bounds__(256) void gcn_hist(const int* __restrict__ u,
                                                int* __restrict__ deg, int E) {
    int e = blockIdx.x * blockDim.x + threadIdx.x;
    if (e < E) atomicAdd(&deg[u[e]], 1);
}

// ---------------------------------------------------------------------------
// CSR build step 2: exclusive prefix scan of deg -> rowptr. Single block of
// 1024 threads, Hillis-Steele per 1024-chunk with a running carry.
// ---------------------------------------------------------------------------
__global__ __launch_bounds__(1024) void gcn_scan(const int* __restrict__ deg,
                                                 int* __restrict__ rowptr, int N) {
    __shared__ int s[1024];
    const int tid = threadIdx.x;
    int carry = 0;
    for (int base = 0; base < N; base += 1024) {
        int d = (base + tid < N) ? deg[base + tid] : 0;
        s[tid] = d;
        __syncthreads();
#pragma unroll
        for (int off = 1; off < 1024; off <<= 1) {
            int t = (tid >= off) ? s[tid - off] : 0;
            __syncthreads();
            s[tid] += t;
            __syncthreads();
        }
        int incl = s[tid];
        int tot  = s[1023];
        if (base + tid < N) rowptr[base + tid] = carry + incl - d;
        carry += tot;
        __syncthreads();
    }
    if (tid == 0) rowptr[N] = carry;
}

// ---------------------------------------------------------------------------
// CSR build step 3a: cursor = rowptr[0..N-1].
// ---------------------------------------------------------------------------
__global__ __launch_bounds__(256) void gcn_copy_i32(const int* __restrict__ src,
                                                    int* __restrict__ dst, int n) {
    int i = blockIdx.x * blockDim.x + threadIdx.x;
    if (i < n) dst[i] = src[i];
}

// ---------------------------------------------------------------------------
// CSR build step 3b: scatter edges into (col, wgt) via per-row cursors.
// ---------------------------------------------------------------------------
__global__ __launch_bounds__(256) void gcn_fill(const int* __restrict__ u,
                                                const int* __restrict__ v,
                                                const float* __restrict__ ew,
                                                int* __restrict__ cursor,
                                                int* __restrict__ col,
                                                float* __restrict__ wgt, int E) {
    int e = blockIdx.x * blockDim.x + threadIdx.x;
    if (e >= E) return;
    int pos = atomicAdd(&cursor[u[e]], 1);
    col[pos] = v[e];
    wgt[pos] = ew[e];
}

// ---------------------------------------------------------------------------
// Transpose W1 (128x128) -> Wt1[c*128+k], W2 (128x16) -> Wt2[c*128+k]
// so the WMMA B operand's per-lane {W[k][n],W[k+1][n]} pair is a contiguous
// 8-byte load.
// ---------------------------------------------------------------------------
__global__ __launch_bounds__(256) void gcn_transpose(const float* __restrict__ W1,
                                                     const float* __restrict__ W2,
                                                     float* __restrict__ Wt1,
                                                     float* __restrict__ Wt2) {
    int t = blockIdx.x * blockDim.x + threadIdx.x;
    if (t < F_IN * F_HID) {
        int k = t >> 7, c = t & 127;
        Wt1[c * F_IN + k] = W1[k * F_HID + c];
    } else {
        int s = t - F_IN * F_HID;
        if (s < F_HID * F_OUT) {
            int k = s >> 4, c = s & 15;
            Wt2[c * F_HID + k] = W2[k * F_OUT + c];
        }
    }
}

// ---------------------------------------------------------------------------
// Gather SpMM (CSR): agg[n] = sum_j wgt[j] * x[col[j]], j in row n.
// One wave32 per node, 4 features per lane in registers, zero FP atomics.
// Rows stream through the 192MB L2; speculative prefetch one iteration ahead
// hides L2->L0 latency on the gather critical path.
// ---------------------------------------------------------------------------
__global__ __launch_bounds__(256) void gcn_spmm_csr(const int* __restrict__ rowptr,
                                                    const int* __restrict__ col,
                                                    const float* __restrict__ wgt,
                                                    const float* __restrict__ x,
                                                    float* __restrict__ agg, int N) {
    int node = blockIdx.x * 8 + (threadIdx.x >> 5);
    if (node >= N) return;
    int lane = threadIdx.x & 31;

    int beg = rowptr[node];
    int end = rowptr[node + 1];

    float ax = 0.f, ay = 0.f, az = 0.f, aw = 0.f;
    int j = beg;
    for (; j + 1 < end; j += 2) {
        int   c0 = col[j],  c1 = col[j + 1];
        float w0 = wgt[j],  w1 = wgt[j + 1];
        if (j + 3 < end) {
            // speculative prefetch of the next pair of gathered rows
            __builtin_prefetch(x + (size_t)col[j + 2] * F_IN + lane * 4, 0, 3);
            __builtin_prefetch(x + (size_t)col[j + 3] * F_IN + lane * 4, 0, 3);
        }
        float4 r0 = ((const float4*)(x + (size_t)c0 * F_IN))[lane];
        float4 r1 = ((const float4*)(x + (size_t)c1 * F_IN))[lane];
        ax = fmaf(w0, r0.x, fmaf(w1, r1.x, ax));
        ay = fmaf(w0, r0.y, fmaf(w1, r1.y, ay));
        az = fmaf(w0, r0.z, fmaf(w1, r1.z, az));
        aw = fmaf(w0, r0.w, fmaf(w1, r1.w, aw));
    }
    if (j < end) {
        int c0 = col[j];
        float w0 = wgt[j];
        float4 r0 = ((const float4*)(x + (size_t)c0 * F_IN))[lane];
        ax = fmaf(w0, r0.x, ax);
        ay = fmaf(w0, r0.y, ay);
        az = fmaf(w0, r0.z, az);
        aw = fmaf(w0, r0.w, aw);
    }
    ((float4*)(agg + (size_t)node * F_IN))[lane] = make_float4(ax, ay, az, aw);
}

// ---------------------------------------------------------------------------
// h = relu(agg @ W1): 64 rows/block, 8 waves. A tile staged into LDS via
// CDNA5 async cache->LDS DMA (no VGPR round-trip, tracked on ASYNCcnt).
// K=128 in steps of 4 via V_WMMA_F32_16X16X4_F32.
//   A lane map (16x4 f32): M = lane&15, K-pair offset kk = 2*(lane>>4)
//   B lane map (4x16 f32): N = lane&15, same kk
//   D lane map (16x16 f32): row = vgpr + 8*(lane>>4), col = lane&15
// ---------------------------------------------------------------------------
__global__ __launch_bounds__(256) void gcn_gemm_relu(const float* __restrict__ agg,
                                                     const float* __restrict__ Wt,
                                                     float* __restrict__ h) {
    __shared__ float As[64 * AS_STRIDE];
    const int block_row0 = blockIdx.x * 64;

    // Stage 64x128 A tile: 2048 float4, 256 threads -> 8 async b128 DMAs each.
#pragma unroll
    for (int i = 0; i < 8; ++i) {
        int idx = threadIdx.x + i * 256;   // 0..2047
        int r = idx >> 5;                  // 32 float4 per row
        int c4 = idx & 31;
        async_copy_b128_to_lds(As + r * AS_STRIDE + c4 * 4,
                               agg + (size_t)(block_row0 + r) * F_IN + c4 * 4);
    }
    wait_async_all();
    __syncthreads();

    const int wave = threadIdx.x >> 5;
    const int lane = threadIdx.x & 31;
    const int hs = lane >> 4;      // half-select
    const int ln = lane & 15;      // M (for A) / N (for B)
    const int kk = hs * 2;         // K-pair offset within step of 4

    const int rt = wave >> 1;          // local row tile 0..3
    const int cb = (wave & 1) * 64;    // col base: 4 col tiles of 16

    const float* arow = As + (rt * 16 + ln) * AS_STRIDE + kk;
    const float* b0p = Wt + (size_t)(cb + 0  + ln) * F_IN + kk;
    const float* b1p = Wt + (size_t)(cb + 16 + ln) * F_IN + kk;
    const float* b2p = Wt + (size_t)(cb + 32 + ln) * F_IN + kk;
    const float* b3p = Wt + (size_t)(cb + 48 + ln) * F_IN + kk;

    v8f c0 = {}, c1 = {}, c2 = {}, c3 = {};
#pragma unroll 8
    for (int k = 0; k < F_IN; k += 4) {
        v2f a = *(const v2f*)(arow + k);   // ds_load (paired by compiler)
        v2f b0 = *(const v2f*)(b0p + k);   // global_load_b64 (L0/L2 resident)
        v2f b1 = *(const v2f*)(b1p + k);
        v2f b2 = *(const v2f*)(b2p + k);
        v2f b3 = *(const v2f*)(b3p + k);
        c0 = wmma_f32_k4(a, b0, c0);
        c1 = wmma_f32_k4(a, b1, c1);
        c2 = wmma_f32_k4(a, b2, c2);
        c3 = wmma_f32_k4(a, b3, c3);
    }

    const int gr = block_row0 + rt * 16 + hs * 8;
#pragma unroll
    for (int i = 0; i < 8; ++i) {
        float* row = h + (size_t)(gr + i) * F_HID + cb + ln;
        row[0]  = fmaxf(c0[i], 0.f);
        row[16] = fmaxf(c1[i], 0.f);
        row[32] = fmaxf(c2[i], 0.f);
        row[48] = fmaxf(c3[i], 0.f);
    }
}

// ---------------------------------------------------------------------------
// out = agg2 @ W2 (128x16). 64 rows/block, 4 waves, one 16x16 tile per wave.
// ---------------------------------------------------------------------------
__global__ __launch_bounds__(128) void gcn_gemm_out(const float* __restrict__ agg,
                                                    const float* __restrict__ Wt2,
                                                    float* __restrict__ out) {
    __shared__ float As[64 * AS_STRIDE];
    const int block_row0 = blockIdx.x * 64;

#pragma unroll
    for (int i = 0; i < 16; ++i) {
        int idx = threadIdx.x + i * 128;   // 0..2047
        int r = idx >> 5;
        int c4 = idx & 31;
        async_copy_b128_to_lds(As + r * AS_STRIDE + c4 * 4,
                               agg + (size_t)(block_row0 + r) * F_HID + c4 * 4);
    }
    wait_async_all();
    __syncthreads();

    const int wave = threadIdx.x >> 5;   // row tile 0..3
    const int lane = threadIdx.x & 31;
    const int hs = lane >> 4;
    const int ln = lane & 15;
    const int kk = hs * 2;

    const float* arow = As + (wave * 16 + ln) * AS_STRIDE + kk;
    const float* bp = Wt2 + (size_t)ln * F_HID + kk;

    v8f c0 = {};
#pragma unroll 8
    for (int k = 0; k < F_HID; k += 4) {
        v2f a = *(const v2f*)(arow + k);
        v2f b = *(const v2f*)(bp + k);
        c0 = wmma_f32_k4(a, b, c0);
    }

    const int gr = block_row0 + wave * 16 + hs * 8;
#pragma unroll
    for (int i = 0; i < 8; ++i) {
        out[(size_t)(gr + i) * F_OUT + ln] = c0[i];
    }
}

// ---------------------------------------------------------------------------
extern "C" void kernel_launch(void* const* d_in, const int* in_sizes, int n_in,
                              void* d_out, int out_size, void* d_ws, size_t ws_size,
                              hipStream_t stream) {
    const float* x  = (const float*)d_in[0];
    const int*   u  = (const int*)d_in[1];
    const int*   v  = (const int*)d_in[2];
    const float* ew = (const float*)d_in[3];
    const float* W1 = (const float*)d_in[4];
    const float* W2 = (const float*)d_in[5];

    const int N = in_sizes[0] / F_IN;
    const int E = in_sizes[1];
    const size_t NF = (size_t)N * F_IN;

    // ws layout (float4-aligned buffers first):
    // [agg | h | wgt | Wt1 | Wt2 | rowptr | cursor | deg | col]
    float* agg  = (float*)d_ws;            // reused for layer 1 and layer 2
    float* h    = agg + NF;
    float* wgt  = h + NF;
    float* Wt1  = wgt + E;
    float* Wt2  = Wt1 + F_IN * F_HID;
    int* rowptr = (int*)(Wt2 + F_HID * F_OUT);
    int* cursor = rowptr + (N + 1);
    int* deg    = cursor + N;
    int* col    = deg + N;

    const int eb = (E + 255) / 256;
    const int nb = (N + 255) / 256;

    // ---- CSR build (per call; deterministic structure) ----
    gcn_zero_i32<<<256, 256, 0, stream>>>(deg, N);
    gcn_hist<<<eb, 256, 0, stream>>>(u, deg, E);
    gcn_scan<<<1, 1024, 0, stream>>>(deg, rowptr, N);
    gcn_copy_i32<<<nb, 256, 0, stream>>>(rowptr, cursor, N);
    gcn_fill<<<eb, 256, 0, stream>>>(u, v, ew, cursor, col, wgt, E);

    // ---- weight transposes for WMMA B-operand layout ----
    {
        int total = F_IN * F_HID + F_HID * F_OUT;
        gcn_transpose<<<(total + 255) / 256, 256, 0, stream>>>(W1, W2, Wt1, Wt2);
    }

    // ---- layer 1 ----
    gcn_spmm_csr<<<(N + 7) / 8, 256, 0, stream>>>(rowptr, col, wgt, x, agg, N);
    gcn_gemm_relu<<<N / 64, 256, 0, stream>>>(agg, Wt1, h);

    // ---- layer 2 (agg buffer reused) ----
    gcn_spmm_csr<<<(N + 7) / 8, 256, 0, stream>>>(rowptr, col, wgt, h, agg, N);
    gcn_gemm_out<<<N / 64, 128, 0, stream>>>(agg, Wt2, (float*)d_out);
}